// Predictor_37563783971320
// MI455X (gfx1250) — compile-verified
//
#include <hip/hip_runtime.h>
#include <hip/hip_bf16.h>

typedef __attribute__((ext_vector_type(2))) float v2f;
typedef __attribute__((ext_vector_type(8))) float v8f;

#define N_NODES 100000
#define N_EDGES 1600000
#define IN_DIM  33
#define HID_DIM 64
#define OUT_DIM 32

// ---------------------------------------------------------------------------
// Degree / normalization
// ---------------------------------------------------------------------------
__global__ void deg_init_kernel(float* __restrict__ deg) {
    int i = blockIdx.x * blockDim.x + threadIdx.x;
    if (i < N_NODES) deg[i] = 1.0f;   // self-loop contributes 1 to in-degree
}

__global__ void deg_accum_kernel(const int* __restrict__ ei, float* __restrict__ deg) {
    int e = blockIdx.x * blockDim.x + threadIdx.x;
    if (e < N_EDGES) {
        int d = ei[N_EDGES + e];      // dst row of edge_index
        atomicAdd(&deg[d], 1.0f);
    }
}

__global__ void deg_rsqrt_kernel(float* __restrict__ deg) {
    int i = blockIdx.x * blockDim.x + threadIdx.x;
    if (i < N_NODES) deg[i] = rsqrtf(deg[i]);   // deg >= 1 always
}

// ---------------------------------------------------------------------------
// GEMM 1: h1[N,64] = concat(z[N,32], action[N,1]) @ W1[33,64]
// One wave -> 16 rows x 64 cols via V_WMMA_F32_16X16X4_F32, K chunked by 4.
// ---------------------------------------------------------------------------
__global__ void gemm1_kernel(const float* __restrict__ z,
                             const float* __restrict__ action,
                             const float* __restrict__ W1,
                             float* __restrict__ h1) {
    const int lane = threadIdx.x & 31;
    const int wave = threadIdx.x >> 5;
    const int m0 = (blockIdx.x * 4 + wave) * 16;
    if (m0 >= N_NODES) return;                 // wave-uniform exit: EXEC stays all-ones

    const int mr = lane & 15;                  // row-in-tile (A) / col-in-tile (B,C,D)
    const int kh = lane >> 4;                  // K-half select for A; M-half for C/D

    int row = m0 + mr;
    if (row >= N_NODES) row = N_NODES - 1;     // clamp loads; stores are guarded

    v8f acc[4] = {v8f{}, v8f{}, v8f{}, v8f{}};

    for (int kc = 0; kc < 9; ++kc) {           // K = 33 padded to 36 = 9 chunks of 4
        const int k0 = kc * 4 + 2 * kh;
        v2f a;
        a.x = (k0     < 32) ? z[row * 32 + k0]
            : ((k0     == 32) ? action[row] : 0.0f);
        a.y = (k0 + 1 < 32) ? z[row * 32 + k0 + 1]
            : ((k0 + 1 == 32) ? action[row] : 0.0f);
#pragma unroll
        for (int nt = 0; nt < 4; ++nt) {
            const int col = nt * 16 + mr;
            v2f b;
            b.x = (k0     < IN_DIM) ? W1[k0 * HID_DIM + col]       : 0.0f;
            b.y = (k0 + 1 < IN_DIM) ? W1[(k0 + 1) * HID_DIM + col] : 0.0f;
            acc[nt] = __builtin_amdgcn_wmma_f32_16x16x4_f32(
                false, a, false, b, (short)0, acc[nt], false, false);
        }
    }

#pragma unroll
    for (int v = 0; v < 8; ++v) {
        const int r = m0 + 8 * kh + v;         // C/D layout: lanes16-31 hold M+8
        if (r < N_NODES) {
#pragma unroll
            for (int nt = 0; nt < 4; ++nt)
                h1[(long long)r * HID_DIM + nt * 16 + mr] = acc[nt][v];
        }
    }
}

// ---------------------------------------------------------------------------
// GEMM 2: h2[N,32] = relu1[N,64] @ W2[64,32]
// ---------------------------------------------------------------------------
__global__ void gemm2_kernel(const float* __restrict__ x,
                             const float* __restrict__ W2,
                             float* __restrict__ h2) {
    const int lane = threadIdx.x & 31;
    const int wave = threadIdx.x >> 5;
    const int m0 = (blockIdx.x * 4 + wave) * 16;
    if (m0 >= N_NODES) return;

    const int mr = lane & 15;
    const int kh = lane >> 4;

    int row = m0 + mr;
    if (row >= N_NODES) row = N_NODES - 1;

    v8f acc[2] = {v8f{}, v8f{}};

    for (int kc = 0; kc < 16; ++kc) {          // K = 64 = 16 chunks of 4, no padding
        const int k0 = kc * 4 + 2 * kh;
        v2f a;
        a.x = x[(long long)row * HID_DIM + k0];
        a.y = x[(long long)row * HID_DIM + k0 + 1];
#pragma unroll
        for (int nt = 0; nt < 2; ++nt) {
            const int col = nt * 16 + mr;
            v2f b;
            b.x = W2[k0 * OUT_DIM + col];
            b.y = W2[(k0 + 1) * OUT_DIM + col];
            acc[nt] = __builtin_amdgcn_wmma_f32_16x16x4_f32(
                false, a, false, b, (short)0, acc[nt], false, false);
        }
    }

#pragma unroll
    for (int v = 0; v < 8; ++v) {
        const int r = m0 + 8 * kh + v;
        if (r < N_NODES) {
#pragma unroll
            for (int nt = 0; nt < 2; ++nt)
                h2[(long long)r * OUT_DIM + nt * 16 + mr] = acc[nt][v];
        }
    }
}

// ---------------------------------------------------------------------------
// Edge aggregation: agg[dst] += h[src] * dinv[src]*dinv[dst]
// One wave per edge; lanes stride the feature dim. Index loads are wave-
// uniform (single L2 broadcast); scatter uses global_atomic_add_f32.
// Working set (~80MB) fits the 192MB L2, so atomics run at L2 bandwidth.
// ---------------------------------------------------------------------------
template <int F>
__global__ void aggregate_kernel(const int* __restrict__ ei,
                                 const float* __restrict__ dinv,
                                 const float* __restrict__ h,
                                 float* __restrict__ agg) {
    const long long t = (long long)blockIdx.x * blockDim.x + threadIdx.x;
    const int lane = threadIdx.x & 31;
    const long long e = t >> 5;
    if (e >= N_EDGES) return;

    const int s = ei[e];
    const int d = ei[N_EDGES + e];
    const float norm = dinv[s] * dinv[d];

    const float* __restrict__ hs = h + (long long)s * F;
    float* __restrict__ ad = agg + (long long)d * F;
#pragma unroll
    for (int f = lane; f < F; f += 32)
        atomicAdd(&ad[f], hs[f] * norm);
}

// ---------------------------------------------------------------------------
// Finalize layer 1: out = relu(agg + h1*dinv^2 (self-loop) + b1), in place
// ---------------------------------------------------------------------------
__global__ void finalize1_kernel(float* __restrict__ agg,
                                 const float* __restrict__ h1,
                                 const float* __restrict__ dinv,
                                 const float* __restrict__ b1) {
    const long long i = (long long)blockIdx.x * blockDim.x + threadIdx.x;
    if (i >= (long long)N_NODES * HID_DIM) return;
    const int node = (int)(i >> 6);
    const int f = (int)(i & 63);
    const float di = dinv[node];
    const float v = agg[i] + h1[i] * di * di + b1[f];
    agg[i] = fmaxf(v, 0.0f);
}

// ---------------------------------------------------------------------------
// Finalize layer 2: out = agg + h2*dinv^2 + b2  (no relu)
// ---------------------------------------------------------------------------
__global__ void finalize2_kernel(const float* __restrict__ agg,
                                 const float* __restrict__ h2,
                                 const float* __restrict__ dinv,
                                 const float* __restrict__ b2,
                                 float* __restrict__ out) {
    const long long i = (long long)blockIdx.x * blockDim.x + threadIdx.x;
    if (i >= (long long)N_NODES * OUT_DIM) return;
    const int node = (int)(i >> 5);
    const int f = (int)(i & 31);
    const float di = dinv[node];
    out[i] = agg[i] + h2[i] * di * di + b2[f];
}

// ---------------------------------------------------------------------------
extern "C" void kernel_launch(void* const* d_in, const int* in_sizes, int n_in,
                              void* d_out, int out_size, void* d_ws, size_t ws_size,
                              hipStream_t stream) {
    const float* z      = (const float*)d_in[0];   // [N,32]
    const float* action = (const float*)d_in[1];   // [N,1]
    const int*   ei     = (const int*)  d_in[2];   // [2,E]
    const float* W1     = (const float*)d_in[3];   // [33,64]
    const float* b1     = (const float*)d_in[4];   // [64]
    const float* W2     = (const float*)d_in[5];   // [64,32]
    const float* b2     = (const float*)d_in[6];   // [32]
    float* out = (float*)d_out;                    // [N,32]

    // Workspace carve-out (floats): dinv | h1 | agg1(->relu1 in place) | h2 | agg2
    float* dinv = (float*)d_ws;
    float* h1   = dinv + N_NODES;
    float* agg1 = h1   + (size_t)N_NODES * HID_DIM;
    float* h2   = agg1 + (size_t)N_NODES * HID_DIM;
    float* agg2 = h2   + (size_t)N_NODES * OUT_DIM;

    const int T = 256;

    // 1) degrees -> dinv
    deg_init_kernel<<<(N_NODES + T - 1) / T, T, 0, stream>>>(dinv);
    deg_accum_kernel<<<(N_EDGES + T - 1) / T, T, 0, stream>>>(ei, dinv);
    deg_rsqrt_kernel<<<(N_NODES + T - 1) / T, T, 0, stream>>>(dinv);

    // 2) h1 = x @ W1  (WMMA f32 16x16x4; 4 waves x 16 rows per block)
    const int gemm_blocks = (N_NODES + 63) / 64;
    gemm1_kernel<<<gemm_blocks, 128, 0, stream>>>(z, action, W1, h1);

    // 3) layer-1 edge aggregation
    hipMemsetAsync(agg1, 0, (size_t)N_NODES * HID_DIM * sizeof(float), stream);
    {
        const long long threads = (long long)N_EDGES * 32;
        const int blocks = (int)((threads + T - 1) / T);
        aggregate_kernel<HID_DIM><<<blocks, T, 0, stream>>>(ei, dinv, h1, agg1);
    }
    finalize1_kernel<<<(int)(((long long)N_NODES * HID_DIM + T - 1) / T), T, 0, stream>>>(
        agg1, h1, dinv, b1);

    // 4) h2 = relu1 @ W2
    gemm2_kernel<<<gemm_blocks, 128, 0, stream>>>(agg1, W2, h2);

    // 5) layer-2 edge aggregation
    hipMemsetAsync(agg2, 0, (size_t)N_NODES * OUT_DIM * sizeof(float), stream);
    {
        const long long threads = (long long)N_EDGES * 32;
        const int blocks = (int)((threads + T - 1) / T);
        aggregate_kernel<OUT_DIM><<<blocks, T, 0, stream>>>(ei, dinv, h2, agg2);
    }
    finalize2_kernel<<<(int)(((long long)N_NODES * OUT_DIM + T - 1) / T), T, 0, stream>>>(
        agg2, h2, dinv, b2, out);
}